// MOLS_62723702391250
// MI455X (gfx1250) — compile-verified
//
#include <hip/hip_runtime.h>
#include <math.h>

// ---------------------------------------------------------------------------
// VMamba cross-scan block for MI455X (gfx1250).
// GEMMs: V_WMMA_F32_16X16X4_F32, template-specialized layouts, 4 N-tiles per
// wave, software-pipelined loads (double buffer). Scan: 16 lanes per channel.
// ---------------------------------------------------------------------------

typedef __attribute__((ext_vector_type(8))) float v8f;
typedef __attribute__((ext_vector_type(2))) float v2f;

#define LSEQ 16384   // H*W = 128*128
#define HH 128
#define NW 8         // waves per GEMM block
#define NT 4         // 16-wide N tiles per wave

// Direction-specific position map (involutive for dirs 0..3).
__device__ __forceinline__ int posmap(int dir, int l) {
    switch (dir) {
        case 0: return l;
        case 1: return (LSEQ - 1) - l;
        case 2: return ((l & (HH - 1)) << 7) | (l >> 7);
        default: {
            int lp = (LSEQ - 1) - l;
            return ((lp & (HH - 1)) << 7) | (lp >> 7);
        }
    }
}

// ---------------------------------------------------------------------------
// Batched f32 WMMA GEMM:  Out = act(W[M x K] * X[K x N])
//   XNK   : 0 -> X[k*ldx + n] (K-major)   1 -> X[n*K + k] (N-major)
//   OUTMM : 0 -> Out[n*ldo + m]           1 -> Out[m*ldo + n]
//   ACT   : 0 none, 1 exact GELU
//   MG    : 1 -> guard rows m >= Mreal (when M % 16 != 0)
// blockDim = (32, NW); each wave owns NT adjacent 16x16 output tiles.
// Loads for step k+1 are issued before the wmma batch of step k.
// ---------------------------------------------------------------------------
template <int XNK, int OUTMM, int ACT, int MG>
__global__ void gemm_wmma_f32(const float* __restrict__ Wm,
                              const float* __restrict__ Xm,
                              float* __restrict__ Om,
                              int Mreal, int K, int ldx, int ldo,
                              unsigned long long wStride, int wShift,
                              unsigned long long xStride,
                              unsigned long long oStride)
{
    const int z = blockIdx.z;
    const float* Wb = Wm + (unsigned long long)(z >> wShift) * wStride;
    const float* Xb = Xm + (unsigned long long)z * xStride;
    float*       Ob = Om + (unsigned long long)z * oStride;

    const int lane  = threadIdx.x;      // 0..31
    const int l16   = lane & 15;
    const int hi    = lane >> 4;        // lane-half: A/B K split {0,1} / {2,3}
    const int kaoff = hi << 1;
    const int n0    = (blockIdx.x * NW + threadIdx.y) * (16 * NT);
    const int m0    = blockIdx.y << 4;

    // A row: clamp + zero-mask instead of divergent branch.
    int   mrow = m0 + l16;
    float msc  = 1.f;
    if (MG) { if (mrow >= Mreal) { mrow = 0; msc = 0.f; } }
    const float* aptr = Wb + (unsigned long long)mrow * K + kaoff;

    v8f acc[NT];
#pragma unroll
    for (int j = 0; j < NT; ++j) acc[j] = (v8f){};

    // B pointers
    const float* bp[NT];                 // N-major per-tile pointers
    const float* b0 = nullptr;           // K-major row pair
    const float* b1 = nullptr;
    if (XNK) {
#pragma unroll
        for (int j = 0; j < NT; ++j)
            bp[j] = Xb + (unsigned long long)(n0 + j * 16 + l16) * K + kaoff;
    } else {
        b0 = Xb + (unsigned long long)kaoff * ldx + n0 + l16;
        b1 = b0 + ldx;
    }

    // ---- prologue: load fragments for k = 0 ----
    v2f a, bv[NT];
    a.x = aptr[0]; a.y = aptr[1];
    if (MG) { a.x *= msc; a.y *= msc; }
    aptr += 4;
#pragma unroll
    for (int j = 0; j < NT; ++j) {
        if (XNK) { bv[j].x = bp[j][0]; bv[j].y = bp[j][1]; bp[j] += 4; }
        else     { bv[j].x = b0[j * 16]; bv[j].y = b1[j * 16]; }
    }
    if (!XNK) { b0 += (unsigned long long)ldx * 4; b1 += (unsigned long long)ldx * 4; }

    // ---- main loop: prefetch k+4, compute k ----
    for (int k0 = 0; k0 < K - 4; k0 += 4) {
        v2f an, bn[NT];
        an.x = aptr[0]; an.y = aptr[1];
        aptr += 4;
#pragma unroll
        for (int j = 0; j < NT; ++j) {
            if (XNK) { bn[j].x = bp[j][0]; bn[j].y = bp[j][1]; bp[j] += 4; }
            else     { bn[j].x = b0[j * 16]; bn[j].y = b1[j * 16]; }
        }
        if (!XNK) { b0 += (unsigned long long)ldx * 4; b1 += (unsigned long long)ldx * 4; }
        if (MG) { an.x *= msc; an.y *= msc; }

#pragma unroll
        for (int j = 0; j < NT; ++j)
            acc[j] = __builtin_amdgcn_wmma_f32_16x16x4_f32(false, a, false, bv[j],
                                                           (short)0, acc[j],
                                                           false, false);
        a = an;
#pragma unroll
        for (int j = 0; j < NT; ++j) bv[j] = bn[j];
    }
    // ---- epilogue: last wmma batch ----
#pragma unroll
    for (int j = 0; j < NT; ++j)
        acc[j] = __builtin_amdgcn_wmma_f32_16x16x4_f32(false, a, false, bv[j],
                                                       (short)0, acc[j],
                                                       false, false);

#pragma unroll
    for (int j = 0; j < NT; ++j) {
        const int n = n0 + j * 16 + l16;
#pragma unroll
        for (int r = 0; r < 8; ++r) {
            int m = m0 + (hi << 3) + r;   // C/D: VGPR r -> M=r (lo) / M=r+8 (hi)
            if (!MG || m < Mreal) {
                float v = acc[j][r];
                if (ACT == 1) v = 0.5f * v * (1.f + erff(v * 0.70710678118f));
                if (OUTMM) Ob[(unsigned long long)m * ldo + n] = v;
                else       Ob[(unsigned long long)n * ldo + m] = v;
            }
        }
    }
}

// ---------------------------------------------------------------------------
// Gather direction sequences from xe[b][pos][208] + LayerNorm over GD=64.
// One thread per (dirb, l).
// ---------------------------------------------------------------------------
__global__ void gather_ln(const float* __restrict__ xe,
                          const float* __restrict__ nw,
                          const float* __restrict__ nb,
                          float* __restrict__ hseq)
{
    int idx = blockIdx.x * blockDim.x + threadIdx.x;
    if (idx >= 8 * LSEQ) return;
    int l    = idx & (LSEQ - 1);
    int dirb = idx >> 14;
    int dir  = dirb >> 1, b = dirb & 1;
    int pos  = posmap(dir, l);
    const float* src = xe + ((unsigned long long)b * LSEQ + pos) * 208 + dir * 48;

    float s = 0.f, ss = 0.f;
#pragma unroll
    for (int g4 = 0; g4 < 16; ++g4) {
        float4 v = *(const float4*)(src + g4 * 4);
        s  += v.x + v.y + v.z + v.w;
        ss += v.x * v.x + v.y * v.y + v.z * v.z + v.w * v.w;
    }
    float mean = s * (1.f / 64.f);
    float var  = ss * (1.f / 64.f) - mean * mean;
    float inv  = rsqrtf(var + 1e-6f);

    float* dst = hseq + (unsigned long long)idx * 64;
    const float* w  = nw + dir * 64;
    const float* bb = nb + dir * 64;
#pragma unroll
    for (int g = 0; g < 64; ++g)
        dst[g] = (src[g] - mean) * inv * w[g] + bb[g];
}

// ---------------------------------------------------------------------------
// Depthwise causal conv (DCONV=4, left pad 3) + bias + SiLU.
// xz layout [dirb][l][256]; xin = cols 0..127. One thread per (dirb,l,ch).
// ---------------------------------------------------------------------------
__global__ void conv_silu(const float* __restrict__ xz,
                          const float* __restrict__ cw,
                          const float* __restrict__ cb,
                          float* __restrict__ u)
{
    int idx = blockIdx.x * blockDim.x + threadIdx.x;
    if (idx >= 8 * LSEQ * 128) return;
    int ch   = idx & 127;
    int l    = (idx >> 7) & (LSEQ - 1);
    int dirb = idx >> 21;
    int dir  = dirb >> 1;

    const float* w = cw + (dir * 128 + ch) * 4;
    float acc = cb[dir * 128 + ch];
    const float* base = xz + (unsigned long long)dirb * LSEQ * 256 + ch;
#pragma unroll
    for (int j = 0; j < 4; ++j) {
        int ll = l - 3 + j;
        if (ll >= 0) acc += w[j] * base[(unsigned long long)ll * 256];
    }
    u[idx] = acc / (1.f + __expf(-acc));   // silu
}

// ---------------------------------------------------------------------------
// Fused dt(softplus) + selective scan + D-skip + SiLU(z) gate. In-place on u.
// 16 lanes per channel (lane = state n). dbl row = [dt_r(4) | B(16) | C(16)]
// padded to 48. block = 512 threads = 32 channels; grid = (4 ch-groups, 8 dirb).
// ---------------------------------------------------------------------------
__global__ void scan_kernel(const float* __restrict__ dbl,
                            float* __restrict__ u_y,
                            const float* __restrict__ xz,
                            const float* __restrict__ dtw,
                            const float* __restrict__ dtb,
                            const float* __restrict__ A_log,
                            const float* __restrict__ Dp)
{
    int t    = threadIdx.x;
    int n    = t & 15;
    int ch   = (t >> 4) + (blockIdx.x << 5);
    int dirb = blockIdx.y;
    int dir  = dirb >> 1;
    int dch  = dir * 128 + ch;

    float4 wdt = *(const float4*)(dtw + dch * 4);
    float  bdt = dtb[dch];
    float  a   = -__expf(A_log[dch * 16 + n]);
    float  dd  = Dp[dch];

    const float* dblp = dbl + (unsigned long long)dirb * LSEQ * 48;
    float*       up   = u_y + (unsigned long long)dirb * LSEQ * 128;
    const float* zp   = xz  + (unsigned long long)dirb * LSEQ * 256 + 128;

    float h = 0.f;
    for (int l = 0; l < LSEQ; ++l) {
        const float* row = dblp + (unsigned long long)l * 48;
        float4 r0 = *(const float4*)row;
        float dtraw = r0.x * wdt.x + r0.y * wdt.y + r0.z * wdt.z + r0.w * wdt.w + bdt;
        float dt = (dtraw > 20.f) ? dtraw : __logf(1.f + __expf(dtraw));  // softplus
        float Bv = row[4 + n];
        float Cv = row[20 + n];
        float uv = up[(unsigned long long)l * 128 + ch];

        h = h * __expf(dt * a) + (dt * uv) * Bv;

        float p = h * Cv;                       // reduce over 16 states
        p += __shfl_xor(p, 1, 32);
        p += __shfl_xor(p, 2, 32);
        p += __shfl_xor(p, 4, 32);
        p += __shfl_xor(p, 8, 32);
        if (n == 0) {
            float zv = zp[(unsigned long long)l * 256 + ch];
            float sz = zv / (1.f + __expf(-zv));
            up[(unsigned long long)l * 128 + ch] = (p + uv * dd) * sz;
        }
    }
}

// ---------------------------------------------------------------------------
// Scatter direction outputs back to spatial ycat[b][p][256] (N-major rows so
// fusion1 can use the fast XNK=1 GEMM path).
// ---------------------------------------------------------------------------
__global__ void scatter_out(const float* __restrict__ outs, float* __restrict__ ycat)
{
    int idx = blockIdx.x * blockDim.x + threadIdx.x;
    if (idx >= 8 * LSEQ * 64) return;
    int g    = idx & 63;
    int l    = (idx >> 6) & (LSEQ - 1);
    int dirb = idx >> 20;
    int dir  = dirb >> 1, b = dirb & 1;
    int pos  = posmap(dir, l);
    ycat[((unsigned long long)b * LSEQ + pos) * 256 + dir * 64 + g] = outs[idx];
}

// ---------------------------------------------------------------------------
extern "C" void kernel_launch(void* const* d_in, const int* in_sizes, int n_in,
                              void* d_out, int out_size, void* d_ws, size_t ws_size,
                              hipStream_t stream)
{
    (void)in_sizes; (void)n_in; (void)out_size; (void)ws_size;
    const float* x         = (const float*)d_in[0];   // (2,192,128,128)
    const float* expand_w  = (const float*)d_in[1];   // (208,192)
    const float* norm_w    = (const float*)d_in[2];   // (4,64)
    const float* norm_b    = (const float*)d_in[3];   // (4,64)
    const float* in_proj_w = (const float*)d_in[4];   // (4,256,64)
    const float* conv_w    = (const float*)d_in[5];   // (4,128,4)
    const float* conv_b    = (const float*)d_in[6];   // (4,128)
    const float* x_proj_w  = (const float*)d_in[7];   // (4,36,128)
    const float* dt_proj_w = (const float*)d_in[8];   // (4,128,4)
    const float* dt_proj_b = (const float*)d_in[9];   // (4,128)
    const float* A_log     = (const float*)d_in[10];  // (4,128,16)
    const float* Dp        = (const float*)d_in[11];  // (4,128)
    const float* out_proj_w= (const float*)d_in[12];  // (4,64,128)
    const float* fusion_w1 = (const float*)d_in[13];  // (384,256)
    const float* fusion_w2 = (const float*)d_in[14];  // (192,384)

    const unsigned long long L = LSEQ;
    float* ws   = (float*)d_ws;
    float* xe   = ws;                       // 2*L*208  (xe[b][p][208])
    float* hseq = xe   + 2ULL * L * 208;    // 8*L*64   (also reused for outs)
    float* xz   = hseq + 8ULL * L * 64;     // 8*L*256  (also reused for h1)
    float* u    = xz   + 8ULL * L * 256;    // 8*L*128  (also yact, in-place scan)
    float* dbl  = u    + 8ULL * L * 128;    // 8*L*48
    float* ycat = dbl  + 8ULL * L * 48;     // 2*L*256  (ycat[b][p][256])

    dim3 gblk(32, NW);
    const int GX = LSEQ / (16 * NT * NW);   // 32 N-blocks

    // 1. expand: xe[b][p][o] = expand_w(208x192) * x[b](192 x L)
    gemm_wmma_f32<0,0,0,0><<<dim3(GX, 13, 2), gblk, 0, stream>>>(
        expand_w, x, xe, 208, 192, LSEQ, 208,
        0ULL, 0, 192ULL * L, L * 208ULL);

    // 2. gather + layernorm -> hseq[dirb][l][64]
    gather_ln<<<dim3((8 * LSEQ) / 256), dim3(256), 0, stream>>>(xe, norm_w, norm_b, hseq);

    // 3. in_proj: xz[dirb][l][256] = in_proj_w[dir] * hseq
    gemm_wmma_f32<1,0,0,0><<<dim3(GX, 16, 8), gblk, 0, stream>>>(
        in_proj_w, hseq, xz, 256, 64, 64, 256,
        256ULL * 64, 1, L * 64ULL, L * 256ULL);

    // 4. causal conv4 + silu -> u[dirb][l][128]
    conv_silu<<<dim3((8 * LSEQ * 128) / 256), dim3(256), 0, stream>>>(xz, conv_w, conv_b, u);

    // 5. x_proj: dbl[dirb][l][48] (36 valid) = x_proj_w[dir] * u
    gemm_wmma_f32<1,0,0,1><<<dim3(GX, 3, 8), gblk, 0, stream>>>(
        x_proj_w, u, dbl, 36, 128, 128, 48,
        36ULL * 128, 1, L * 128ULL, L * 48ULL);

    // 6. fused dt + selective scan + gate (in-place: u -> y)
    scan_kernel<<<dim3(4, 8), dim3(512), 0, stream>>>(dbl, u, xz,
                                                      dt_proj_w, dt_proj_b, A_log, Dp);

    // 7. out_proj: outs[dirb][l][64] = out_proj_w[dir] * y   (reuse hseq buffer)
    gemm_wmma_f32<1,0,0,0><<<dim3(GX, 4, 8), gblk, 0, stream>>>(
        out_proj_w, u, hseq, 64, 128, 128, 64,
        64ULL * 128, 1, L * 128ULL, L * 64ULL);

    // 8. scatter -> ycat[b][p][256]
    scatter_out<<<dim3((8 * LSEQ * 64) / 256), dim3(256), 0, stream>>>(hseq, ycat);

    // 9. fusion1 + gelu: h1[b][p][384]  (reuse xz buffer; ycat is N-major)
    gemm_wmma_f32<1,0,1,0><<<dim3(GX, 24, 2), gblk, 0, stream>>>(
        fusion_w1, ycat, xz, 384, 256, 256, 384,
        0ULL, 0, 256ULL * L, L * 384ULL);

    // 10. fusion2 -> d_out[b][192][p]
    gemm_wmma_f32<1,1,0,0><<<dim3(GX, 12, 2), gblk, 0, stream>>>(
        fusion_w2, xz, (float*)d_out, 192, 384, 384, LSEQ,
        0ULL, 0, 384ULL * L, 192ULL * L);
}